// ScamDetectionGNN_2199023255949
// MI455X (gfx1250) — compile-verified
//
#include <hip/hip_runtime.h>
#include <hip/hip_bf16.h>

#define IN_CH  64
#define HID_CH 64
#define OUT_CH 16

typedef __attribute__((ext_vector_type(2))) float v2f;
typedef __attribute__((ext_vector_type(8))) float v8f;

// ---------------- utility ----------------
__global__ void zero_u32_kernel(unsigned* __restrict__ p, long long n) {
  long long i = (long long)blockIdx.x * blockDim.x + threadIdx.x;
  if (i < n) p[i] = 0u;
}

__global__ void degree_kernel(const int* __restrict__ dst, unsigned* __restrict__ deg, int E) {
  int e = blockIdx.x * blockDim.x + threadIdx.x;
  if (e < E) atomicAdd(&deg[dst[e]], 1u);
}

// In-place: buffer holds u32 degree counts, overwritten with float rsqrt(deg+1).
__global__ void dinv_kernel(float* __restrict__ degdinv, int N) {
  int i = blockIdx.x * blockDim.x + threadIdx.x;
  if (i < N) {
    unsigned d = ((const unsigned*)degdinv)[i];
    degdinv[i] = rsqrtf((float)(d + 1u));   // +1 = self loop
  }
}

// ---------------- fp32 WMMA GEMM: H[N,NOUT] = X[N,64] @ W[64,NOUT] ----------------
// One wave computes a 16 x NOUT tile using V_WMMA_F32_16X16X4_F32 (16 K-steps of 4).
template<int NOUT>
__global__ __launch_bounds__(256) void gcn_gemm_wmma(const float* __restrict__ X,
                                                     const float* __restrict__ W,
                                                     float* __restrict__ H,
                                                     int nTiles) {
  const int lane = threadIdx.x & 31;
  const int wave = threadIdx.x >> 5;
  const int tile = blockIdx.x * (blockDim.x >> 5) + wave;
  if (tile >= nTiles) return;                  // wave-uniform branch (EXEC stays all-1)
  const int rowBase = tile * 16;
  const int m  = lane & 15;                    // row within tile (A) / col within ntile (B,C)
  const int kh = lane >> 4;                    // K-half select (0 or 1)

  const float* xrow = X + (size_t)(rowBase + m) * 64;

  constexpr int NT = NOUT / 16;
  v8f acc[NT];
  v8f zero = {};
#pragma unroll
  for (int nt = 0; nt < NT; ++nt) acc[nt] = zero;

#pragma unroll
  for (int kt = 0; kt < 16; ++kt) {
    const int kk = kt * 4 + kh * 2;
    // A-frag: lane holds X[rowBase+m][kk], X[rowBase+m][kk+1]  (contiguous -> b64 load)
    v2f a = *(const v2f*)(xrow + kk);
#pragma unroll
    for (int nt = 0; nt < NT; ++nt) {
      // B-frag: lane holds W[kk][nt*16+m], W[kk+1][nt*16+m]
      v2f b;
      b.x = W[(size_t)kk * NOUT + nt * 16 + m];
      b.y = W[(size_t)(kk + 1) * NOUT + nt * 16 + m];
      acc[nt] = __builtin_amdgcn_wmma_f32_16x16x4_f32(
          /*neg_a=*/false, a, /*neg_b=*/false, b,
          /*c_mod=*/(short)0, acc[nt], /*reuse_a=*/false, /*reuse_b=*/false);
    }
  }

  // C/D layout: VGPR v -> row = rowBase + v + 8*kh, col = nt*16 + m
#pragma unroll
  for (int nt = 0; nt < NT; ++nt) {
#pragma unroll
    for (int v = 0; v < 8; ++v) {
      H[(size_t)(rowBase + v + 8 * kh) * NOUT + nt * 16 + m] = acc[nt][v];
    }
  }
}

// Scalar tail for rows not covered by full 16-row tiles (N=100000 -> never launched).
template<int NOUT>
__global__ void gcn_gemm_tail(const float* __restrict__ X, const float* __restrict__ W,
                              float* __restrict__ H, int rowStart, int N) {
  int idx = blockIdx.x * blockDim.x + threadIdx.x;
  int row = rowStart + idx / NOUT;
  int col = idx % NOUT;
  if (row >= N) return;
  float s = 0.f;
  for (int k = 0; k < 64; ++k) s += X[(size_t)row * 64 + k] * W[(size_t)k * NOUT + col];
  H[(size_t)row * NOUT + col] = s;
}

// ---------------- edge scatter: out[dst] += H[src] * dinv[src]*dinv[dst] ----------------
// One thread per (edge, 4-channel group); float4 gather (L2-resident) + f32 atomics.
template<int C>
__global__ void scatter_kernel(const float* __restrict__ H,
                               const int* __restrict__ src, const int* __restrict__ dst,
                               const float* __restrict__ dinv,
                               float* __restrict__ out, int E) {
  constexpr int G = C / 4;
  long long t = (long long)blockIdx.x * blockDim.x + threadIdx.x;
  if (t >= (long long)E * G) return;
  int e = (int)(t / G);
  int g = (int)(t % G);
  int s = src[e], d = dst[e];
  float nrm = dinv[s] * dinv[d];
  const float4 v = *(const float4*)(H + (size_t)s * C + g * 4);
  float* o = out + (size_t)d * C + g * 4;
  atomicAdd(o + 0, v.x * nrm);
  atomicAdd(o + 1, v.y * nrm);
  atomicAdd(o + 2, v.z * nrm);
  atomicAdd(o + 3, v.w * nrm);
}

// Layer-1 epilogue: add analytic self-loop + bias, ReLU (in place on agg).
__global__ void relu_self_bias_kernel(float* __restrict__ agg, const float* __restrict__ h,
                                      const float* __restrict__ dinv,
                                      const float* __restrict__ b, int N) {
  long long t = (long long)blockIdx.x * blockDim.x + threadIdx.x;
  if (t >= (long long)N * HID_CH) return;
  int i = (int)(t >> 6);
  int c = (int)(t & 63);
  float di = dinv[i];
  float v = agg[t] + h[t] * di * di + b[c];
  agg[t] = fmaxf(v, 0.f);
}

// Layer-2 epilogue: out += self-loop + bias.
__global__ void out_self_bias_kernel(float* __restrict__ out, const float* __restrict__ h2,
                                     const float* __restrict__ dinv,
                                     const float* __restrict__ b, int N) {
  long long t = (long long)blockIdx.x * blockDim.x + threadIdx.x;
  if (t >= (long long)N * OUT_CH) return;
  int i = (int)(t >> 4);
  int c = (int)(t & 15);
  float di = dinv[i];
  out[t] += h2[t] * di * di + b[c];
}

// ---------------- launcher ----------------
extern "C" void kernel_launch(void* const* d_in, const int* in_sizes, int n_in,
                              void* d_out, int out_size, void* d_ws, size_t ws_size,
                              hipStream_t stream) {
  (void)n_in; (void)out_size; (void)ws_size;
  const float* x  = (const float*)d_in[0];
  const int*   ei = (const int*)d_in[1];
  const float* W1 = (const float*)d_in[2];
  const float* b1 = (const float*)d_in[3];
  const float* W2 = (const float*)d_in[4];
  const float* b2 = (const float*)d_in[5];
  const int N = in_sizes[0] / IN_CH;
  const int E = in_sizes[1] / 2;
  const int* src = ei;
  const int* dst = ei + E;
  float* out = (float*)d_out;

  // workspace carve-out (256B aligned): dinv | h1 | agg1 | h2  (~58 MB)
  char* ws = (char*)d_ws;
  size_t off = 0;
  auto carve = [&](size_t bytes) -> void* {
    void* p = ws + off;
    off = (off + bytes + 255) & ~(size_t)255;
    return p;
  };
  float* dinv = (float*)carve((size_t)N * 4);
  float* h1   = (float*)carve((size_t)N * HID_CH * 4);
  float* agg1 = (float*)carve((size_t)N * HID_CH * 4);
  float* h2   = (float*)carve((size_t)N * OUT_CH * 4);

  const int T = 256;
  auto nb = [&](long long n) -> unsigned { return (unsigned)((n + T - 1) / T); };

  // zero accumulators (harness poisons with 0xAA; must re-zero every call)
  zero_u32_kernel<<<nb(N), T, 0, stream>>>((unsigned*)dinv, N);
  zero_u32_kernel<<<nb((long long)N * HID_CH), T, 0, stream>>>((unsigned*)agg1, (long long)N * HID_CH);
  zero_u32_kernel<<<nb((long long)N * OUT_CH), T, 0, stream>>>((unsigned*)out, (long long)N * OUT_CH);

  // degrees (with self loop) -> dinv = rsqrt(deg)
  degree_kernel<<<nb(E), T, 0, stream>>>(dst, (unsigned*)dinv, E);
  dinv_kernel<<<nb(N), T, 0, stream>>>(dinv, N);

  const int nTiles = N / 16;
  const int tailRows = N - nTiles * 16;

  // layer 1: h1 = x @ W1 (fp32 WMMA)
  if (nTiles > 0)
    gcn_gemm_wmma<HID_CH><<<(nTiles + 7) / 8, 256, 0, stream>>>(x, W1, h1, nTiles);
  if (tailRows > 0)
    gcn_gemm_tail<HID_CH><<<nb((long long)tailRows * HID_CH), T, 0, stream>>>(x, W1, h1, nTiles * 16, N);

  // layer 1 aggregation over edges, then self-loop + bias + ReLU
  scatter_kernel<HID_CH><<<nb((long long)E * (HID_CH / 4)), T, 0, stream>>>(h1, src, dst, dinv, agg1, E);
  relu_self_bias_kernel<<<nb((long long)N * HID_CH), T, 0, stream>>>(agg1, h1, dinv, b1, N);

  // layer 2: h2 = relu(agg1) @ W2 (fp32 WMMA)
  if (nTiles > 0)
    gcn_gemm_wmma<OUT_CH><<<(nTiles + 7) / 8, 256, 0, stream>>>(agg1, W2, h2, nTiles);
  if (tailRows > 0)
    gcn_gemm_tail<OUT_CH><<<nb((long long)tailRows * OUT_CH), T, 0, stream>>>(agg1, W2, h2, nTiles * 16, N);

  // layer 2 aggregation directly into d_out, then self-loop + bias
  scatter_kernel<OUT_CH><<<nb((long long)E * (OUT_CH / 4)), T, 0, stream>>>(h2, src, dst, dinv, out, E);
  out_self_bias_kernel<<<nb((long long)N * OUT_CH), T, 0, stream>>>(out, h2, dinv, b2, N);
}